// DecoderWithMask_89180700934286
// MI455X (gfx1250) — compile-verified
//
#include <hip/hip_runtime.h>
#include <math.h>
#include <stdint.h>

// MI455X (gfx1250) fused decoder step.
// - Dominant GEMM (enc_out x U_w^T, 137 GFLOP fp32) on V_WMMA_F32_16X16X4_F32,
//   tanh/v-dot epilogue fused so the 256MB "uh" tensor never exists.
// - enc tiles staged LDS via the Tensor Data Mover (tensor_load_to_lds),
//   double-buffered so TDM DMA overlaps WMMA compute (TENSORcnt pipeline).
//   TDM hardware padding (4 DWORDs per 256) reproduces the 260-float LDS
//   stride that keeps half-wave fragment reads bank-conflict free.
// - logits GEMM streams Wout (128MB) exactly once from HBM with NT loads.

typedef float v2f __attribute__((ext_vector_type(2)));
typedef float v4f __attribute__((ext_vector_type(4)));
typedef float v8f __attribute__((ext_vector_type(8)));
typedef unsigned int u32x4 __attribute__((ext_vector_type(4)));
typedef int i32x4 __attribute__((ext_vector_type(4)));
typedef int i32x8 __attribute__((ext_vector_type(8)));

#define S_LEN 1024
#define BATCH 64
#define EDIM  1024
#define HDIM  1024
#define ADIM  1024
#define EMBD  512
#define VOCAB 32000

// workspace layout (float offsets)
#define WS_WQ      0
#define WS_SCORES  (WS_WQ     + BATCH*ADIM)
#define WS_CTXP    (WS_SCORES + BATCH*S_LEN)
#define WS_CTX     (WS_CTXP   + 8*BATCH*EDIM)
#define WS_X       (WS_CTX    + BATCH*EDIM)
#define WS_GH      (WS_X      + BATCH*HDIM)
#define WS_GI      (WS_GH     + BATCH*3*HDIM)
#define WS_HNEW    (WS_GI     + BATCH*3*HDIM)
#define WS_LOGITS  (WS_HNEW   + BATCH*HDIM)

// output layout (float offsets): [dec_output B*V][h_new B*H][attn B*S]
#define OUT_H    (BATCH*VOCAB)
#define OUT_ATTN (BATCH*VOCAB + BATCH*HDIM)

// ---------------------------------------------------------------------------
// TDM: DMA one 16-row x 256-float fp32 tile (row stride EDIM in memory) into
// LDS at lds_off, with 4-DWORD padding after every 256 DWORDs (-> stride 260).
// D# fields per CDNA5 ISA ch.8: group0 {count=1, lds_addr, global_addr,
// type=2}; group1 {data_size=4B, pad_enable, pad_interval=256dw,
// pad_amount=4dw, tensor=256x16, tile=256x16, dim0_stride=1024}.
// ---------------------------------------------------------------------------
__device__ __forceinline__ void tdm_load_tile16x256(unsigned lds_off,
                                                    unsigned long long ga) {
  u32x4 g0;
  g0[0] = 1u;                                              // count=1, user D#
  g0[1] = lds_off;                                         // LDS byte address
  g0[2] = (unsigned)ga;                                    // global_addr[31:0]
  g0[3] = (unsigned)((ga >> 32) & 0x01FFFFFFull) | (2u << 30); // [56:32]|type=2
  i32x8 g1;
  g1[0] = (2 << 16) | (1 << 20) | (7 << 22) | (3 << 25);   // dsz=4B,pad 4/256dw
  g1[1] = (int)(256u << 16);                               // tensor_dim0=256
  g1[2] = (int)(16u << 16);                                // tensor_dim1=16
  g1[3] = (int)(256u << 16);                               // tile_dim0=256
  g1[4] = 16;                                              // tile_dim1=16
  g1[5] = EDIM;                                            // dim0_stride=1024
  g1[6] = 0;
  g1[7] = 0;
  const i32x4 gz4 = {0, 0, 0, 0};                          // <=2D: groups 2/3
  const i32x8 gz8 = {0, 0, 0, 0, 0, 0, 0, 0};
  __builtin_amdgcn_tensor_load_to_lds(g0, g1, gz4, gz4, gz8, 0);
}

// ---------------------------------------------------------------------------
// Generic row-major linear: out[r*C+c] = dot(in[r,0:K], W[c,0:K]) + bias[c]
// ---------------------------------------------------------------------------
__global__ __launch_bounds__(256) void linear_kernel(
    const float* __restrict__ in, const float* __restrict__ W,
    const float* __restrict__ bias, float* __restrict__ out,
    int C, int K) {
  const int o = blockIdx.x * 256 + threadIdx.x;
  const int r = o / C, c = o % C;
  const float4* a = (const float4*)(in + (size_t)r * K);
  const float4* w = (const float4*)(W + (size_t)c * K);
  float acc = 0.f;
  const int k4 = K >> 2;
  for (int k = 0; k < k4; ++k) {
    const float4 av = a[k], wv = w[k];
    acc += av.x * wv.x + av.y * wv.y + av.z * wv.z + av.w * wv.w;
  }
  out[o] = acc + bias[c];
}

// ---------------------------------------------------------------------------
// Attention scores, fully fused:
//   scores[b][s] = mask(s) * ( v_b + sum_a v_w[a]*tanh(uh[s,b,a]+wq[b,a]+U_b[a]) )
// Workgroup = (one s, 16 batch rows). 8 waves; wave w owns N-tiles [8w,8w+8).
// enc rows DMA'd by TDM into double-buffered LDS chunks; WMMA f32 16x16x4.
// ---------------------------------------------------------------------------
__global__ __launch_bounds__(256) void scores_wmma_kernel(
    const float* __restrict__ enc, const float* __restrict__ Uw,
    const float* __restrict__ Ub, const float* __restrict__ vw,
    const float* __restrict__ vb, const float* __restrict__ wq,
    const int* __restrict__ maskp, float* __restrict__ scores) {
  __shared__ float encT[2][16 * 260];
  __shared__ float wavePart[8][16];

  const int s    = blockIdx.x >> 2;
  const int b0   = (blockIdx.x & 3) << 4;
  const int t    = threadIdx.x;
  const int wave = t >> 5;
  const int lane = t & 31;
  const int ln   = lane & 15;            // row-within-half / N column
  const int kl   = (lane >> 4) << 1;     // K offset of this half-wave (0 or 2)
  const int mhi  = (lane >= 16) ? 8 : 0; // C/D: lanes 16-31 hold M+8

  const unsigned long long gbase =
      (unsigned long long)(uintptr_t)enc +
      ((unsigned long long)s * BATCH * EDIM + (unsigned long long)b0 * EDIM) * 4ull;
  const unsigned lds0 = (unsigned)(uintptr_t)(void*)&encT[0][0]; // low 32b = LDS offset
  const unsigned lds1 = (unsigned)(uintptr_t)(void*)&encT[1][0];

  v8f acc[8];
#pragma unroll
  for (int i = 0; i < 8; ++i)
#pragma unroll
    for (int j = 0; j < 8; ++j) acc[i][j] = 0.f;

  if (wave == 0) tdm_load_tile16x256(lds0, gbase);         // prefetch chunk 0

  for (int kc = 0; kc < 4; ++kc) {
    const int kbase = kc << 8;
    if (wave == 0) {
      if (kc < 3) {  // issue next chunk, then wait for current (in-order TDM)
        tdm_load_tile16x256((kc & 1) ? lds0 : lds1,
                            gbase + (unsigned long long)((kc + 1) << 8) * 4ull);
        __builtin_amdgcn_s_wait_tensorcnt(1);
      } else {
        __builtin_amdgcn_s_wait_tensorcnt(0);
      }
    }
    __syncthreads();

    const float* encbuf = &encT[kc & 1][0];
#pragma unroll
    for (int ni = 0; ni < 8; ++ni) {
      const int n0 = ((wave << 3) + ni) << 4;
      const float* brow = Uw + (size_t)(n0 + ln) * EDIM + kbase + kl;
      const float* arow = encbuf + ln * 260 + kl;
      for (int kk = 0; kk < 64; ++kk) {
        const v2f a = *(const v2f*)(arow + (kk << 2));
        const v2f b = *(const v2f*)(brow + (kk << 2));
        acc[ni] = __builtin_amdgcn_wmma_f32_16x16x4_f32(
            false, a, false, b, (short)0, acc[ni], false, false);
      }
    }
    __syncthreads();  // all waves done with this buffer before TDM reuses it
  }

  // epilogue: tanh(uh + wq + U_b) * v_w, reduce over A
  float rowsum[8];
#pragma unroll
  for (int i = 0; i < 8; ++i) rowsum[i] = 0.f;
#pragma unroll
  for (int ni = 0; ni < 8; ++ni) {
    const int n = (((wave << 3) + ni) << 4) + ln;
    const float ubn = Ub[n], vwn = vw[n];
#pragma unroll
    for (int i = 0; i < 8; ++i) {
      const int m = i + mhi;
      const float u = acc[ni][i] + wq[(size_t)(b0 + m) * ADIM + n] + ubn;
      rowsum[i] += tanhf(u) * vwn;
    }
  }
  // reduce the 16 N-lanes of each half-wave
#pragma unroll
  for (int i = 0; i < 8; ++i) {
    rowsum[i] += __shfl_xor(rowsum[i], 1);
    rowsum[i] += __shfl_xor(rowsum[i], 2);
    rowsum[i] += __shfl_xor(rowsum[i], 4);
    rowsum[i] += __shfl_xor(rowsum[i], 8);
  }
  if (ln == 0) {
#pragma unroll
    for (int i = 0; i < 8; ++i) wavePart[wave][mhi + i] = rowsum[i];
  }
  __syncthreads();
  if (t < 16) {
    float sum = 0.f;
#pragma unroll
    for (int w = 0; w < 8; ++w) sum += wavePart[w][t];
    int m = maskp[0];
    if (m > S_LEN) m = S_LEN;
    const float val = (s < m) ? (sum + vb[0]) : 0.f;  // multiplicative mask
    scores[(size_t)(b0 + t) * S_LEN + s] = val;
  }
}

// ---------------------------------------------------------------------------
// softmax over S per batch row -> attn (written straight to output)
// ---------------------------------------------------------------------------
__global__ __launch_bounds__(256) void softmax_kernel(
    const float* __restrict__ scores, float* __restrict__ out_attn) {
  __shared__ float red[256];
  const int b = blockIdx.x, t = threadIdx.x;
  const float* row = scores + (size_t)b * S_LEN;
  float mx = -1e30f;
  for (int s = t; s < S_LEN; s += 256) mx = fmaxf(mx, row[s]);
  red[t] = mx; __syncthreads();
  for (int o = 128; o > 0; o >>= 1) { if (t < o) red[t] = fmaxf(red[t], red[t + o]); __syncthreads(); }
  mx = red[0]; __syncthreads();
  float sum = 0.f;
  for (int s = t; s < S_LEN; s += 256) sum += expf(row[s] - mx);
  red[t] = sum; __syncthreads();
  for (int o = 128; o > 0; o >>= 1) { if (t < o) red[t] += red[t + o]; __syncthreads(); }
  const float inv = 1.f / red[0];
  for (int s = t; s < S_LEN; s += 256)
    out_attn[(size_t)b * S_LEN + s] = expf(row[s] - mx) * inv;
}

// ---------------------------------------------------------------------------
// ctx[b][e] = sum_s attn[b][s] * enc[s][b][e]  (streaming; NT loads, 2-stage)
// ---------------------------------------------------------------------------
__global__ __launch_bounds__(256) void ctx_partial_kernel(
    const float* __restrict__ attn, const float* __restrict__ enc,
    float* __restrict__ part) {
  const int b = blockIdx.x, sc = blockIdx.y, t = threadIdx.x;
  const int e = t << 2;
  v4f acc = {0.f, 0.f, 0.f, 0.f};
  const int s0 = sc << 7;
  for (int si = 0; si < 128; ++si) {
    const int s = s0 + si;
    const float a = attn[(size_t)b * S_LEN + s];
    const v4f ev = __builtin_nontemporal_load(
        (const v4f*)(enc + (size_t)s * BATCH * EDIM + (size_t)b * EDIM + e));
    acc += a * ev;
  }
  *(v4f*)(part + ((size_t)sc * BATCH + b) * EDIM + e) = acc;
}

__global__ __launch_bounds__(256) void ctx_reduce_kernel(
    const float* __restrict__ part, float* __restrict__ ctx) {
  const int o = blockIdx.x * 256 + threadIdx.x;       // float4 index
  const size_t base = (size_t)o * 4;
  v4f sum = {0.f, 0.f, 0.f, 0.f};
#pragma unroll
  for (int sc = 0; sc < 8; ++sc)
    sum += *(const v4f*)(part + (size_t)sc * BATCH * EDIM + base);
  *(v4f*)(ctx + base) = sum;
}

// ---------------------------------------------------------------------------
// x[b][h] = dot(concat(prev_y, ctx)[b], Wc[h]) + Wc_b[h]  (no concat buffer)
// ---------------------------------------------------------------------------
__global__ __launch_bounds__(256) void combine_kernel(
    const float* __restrict__ prev_y, const float* __restrict__ ctx,
    const float* __restrict__ Wc, const float* __restrict__ Wcb,
    float* __restrict__ x) {
  const int o = blockIdx.x * 256 + threadIdx.x;
  const int b = o >> 10, hh = o & 1023;
  const float4* a1 = (const float4*)(prev_y + (size_t)b * EMBD);
  const float4* a2 = (const float4*)(ctx + (size_t)b * EDIM);
  const float4* w1 = (const float4*)(Wc + (size_t)hh * (EMBD + EDIM));
  const float4* w2 = w1 + (EMBD >> 2);
  float acc = 0.f;
  for (int k = 0; k < (EMBD >> 2); ++k) {
    const float4 av = a1[k], wv = w1[k];
    acc += av.x * wv.x + av.y * wv.y + av.z * wv.z + av.w * wv.w;
  }
  for (int k = 0; k < (EDIM >> 2); ++k) {
    const float4 av = a2[k], wv = w2[k];
    acc += av.x * wv.x + av.y * wv.y + av.z * wv.z + av.w * wv.w;
  }
  x[o] = acc + Wcb[hh];
}

// ---------------------------------------------------------------------------
// GRU gates (PyTorch order r,z,n) -> h_new  (ws + output section)
// ---------------------------------------------------------------------------
__global__ __launch_bounds__(256) void gate_kernel(
    const float* __restrict__ gi, const float* __restrict__ gh,
    const float* __restrict__ hprev, float* __restrict__ hnew_ws,
    float* __restrict__ out_h) {
  const int o = blockIdx.x * 256 + threadIdx.x;
  const int b = o >> 10, h = o & 1023;
  const size_t base = (size_t)b * 3 * HDIM;
  const float r = 1.f / (1.f + expf(-(gi[base + h] + gh[base + h])));
  const float z = 1.f / (1.f + expf(-(gi[base + HDIM + h] + gh[base + HDIM + h])));
  const float n = tanhf(gi[base + 2 * HDIM + h] + r * gh[base + 2 * HDIM + h]);
  const float hp = hprev[o];
  const float hn = (1.f - z) * n + z * hp;
  hnew_ws[o] = hn;
  out_h[o] = hn;
}

// ---------------------------------------------------------------------------
// logits[64 x 32000] = h_new @ Wout^T + Wout_b.  One wave per 16-col V tile,
// all four 16-row M subtiles accumulated so Wout streams from HBM exactly
// once; NT loads keep the single-use 128MB stream out of L2.
// ---------------------------------------------------------------------------
__global__ __launch_bounds__(32) void logits_wmma_kernel(
    const float* __restrict__ hnew, const float* __restrict__ Wout,
    const float* __restrict__ Woutb, float* __restrict__ logits) {
  const int lane = threadIdx.x & 31;
  const int ln   = lane & 15;
  const int kl   = (lane >> 4) << 1;
  const int mhi  = (lane >= 16) ? 8 : 0;
  const int n0   = blockIdx.x << 4;

  v8f acc[4];
#pragma unroll
  for (int i = 0; i < 4; ++i)
#pragma unroll
    for (int j = 0; j < 8; ++j) acc[i][j] = 0.f;

  const float* brow = Wout + (size_t)(n0 + ln) * HDIM + kl;
  for (int k4 = 0; k4 < HDIM; k4 += 4) {
    const v2f b = __builtin_nontemporal_load((const v2f*)(brow + k4));
#pragma unroll
    for (int ms = 0; ms < 4; ++ms) {
      const v2f a = *(const v2f*)(hnew + (size_t)((ms << 4) + ln) * HDIM + kl + k4);
      acc[ms] = __builtin_amdgcn_wmma_f32_16x16x4_f32(
          false, a, false, b, (short)0, acc[ms], false, false);
    }
  }
  const int v = n0 + ln;
  const float wb = Woutb[v];
#pragma unroll
  for (int ms = 0; ms < 4; ++ms)
#pragma unroll
    for (int i = 0; i < 8; ++i) {
      const int b = (ms << 4) + i + mhi;
      logits[(size_t)b * VOCAB + v] = acc[ms][i] + wb;
    }
}

// ---------------------------------------------------------------------------
// log_softmax over V per batch row
// ---------------------------------------------------------------------------
__global__ __launch_bounds__(256) void logsm_kernel(
    const float* __restrict__ logits, float* __restrict__ out) {
  __shared__ float red[256];
  const int b = blockIdx.x, t = threadIdx.x;
  const float* row = logits + (size_t)b * VOCAB;
  float mx = -1e30f;
  for (int v = t; v < VOCAB; v += 256) mx = fmaxf(mx, row[v]);
  red[t] = mx; __syncthreads();
  for (int o = 128; o > 0; o >>= 1) { if (t < o) red[t] = fmaxf(red[t], red[t + o]); __syncthreads(); }
  mx = red[0]; __syncthreads();
  float sum = 0.f;
  for (int v = t; v < VOCAB; v += 256) sum += expf(row[v] - mx);
  red[t] = sum; __syncthreads();
  for (int o = 128; o > 0; o >>= 1) { if (t < o) red[t] += red[t + o]; __syncthreads(); }
  const float lse = mx + logf(red[0]);
  for (int v = t; v < VOCAB; v += 256)
    out[(size_t)b * VOCAB + v] = row[v] - lse;
}

// ---------------------------------------------------------------------------
extern "C" void kernel_launch(void* const* d_in, const int* in_sizes, int n_in,
                              void* d_out, int out_size, void* d_ws, size_t ws_size,
                              hipStream_t stream) {
  (void)in_sizes; (void)n_in; (void)out_size; (void)ws_size;
  const float* prev_y = (const float*)d_in[0];
  const float* prev_h = (const float*)d_in[1];   // [1,B,H] == [B,H]
  const float* enc    = (const float*)d_in[2];
  const float* U_w    = (const float*)d_in[3];
  const float* U_b    = (const float*)d_in[4];
  const float* W_w    = (const float*)d_in[5];
  const float* W_b    = (const float*)d_in[6];
  const float* v_w    = (const float*)d_in[7];
  const float* v_b    = (const float*)d_in[8];
  const float* Wc_w   = (const float*)d_in[9];
  const float* Wc_b   = (const float*)d_in[10];
  const float* W_ih   = (const float*)d_in[11];
  const float* W_hh   = (const float*)d_in[12];
  const float* b_ih   = (const float*)d_in[13];
  const float* b_hh   = (const float*)d_in[14];
  const float* Wout_w = (const float*)d_in[15];
  const float* Wout_b = (const float*)d_in[16];
  const int*   maskp  = (const int*)d_in[17];

  float* out = (float*)d_out;
  float* ws  = (float*)d_ws;
  float* wq     = ws + WS_WQ;
  float* scores = ws + WS_SCORES;
  float* ctxp   = ws + WS_CTXP;
  float* ctx    = ws + WS_CTX;
  float* x      = ws + WS_X;
  float* gh     = ws + WS_GH;
  float* gi     = ws + WS_GI;
  float* hnew   = ws + WS_HNEW;
  float* logits = ws + WS_LOGITS;

  // wq = h @ W_w^T + W_b ; gh = h @ W_hh^T + b_hh (both independent of attn)
  linear_kernel<<<(BATCH * ADIM) / 256, 256, 0, stream>>>(prev_h, W_w, W_b, wq, ADIM, HDIM);
  linear_kernel<<<(BATCH * 3 * HDIM) / 256, 256, 0, stream>>>(prev_h, W_hh, b_hh, gh, 3 * HDIM, HDIM);

  // fused WMMA attention scores (masked), TDM-staged enc tiles
  scores_wmma_kernel<<<S_LEN * (BATCH / 16), 256, 0, stream>>>(
      enc, U_w, U_b, v_w, v_b, wq, maskp, scores);

  // softmax -> attn (output section 3)
  softmax_kernel<<<BATCH, 256, 0, stream>>>(scores, out + OUT_ATTN);

  // context vector
  ctx_partial_kernel<<<dim3(BATCH, 8), 256, 0, stream>>>(out + OUT_ATTN, enc, ctxp);
  ctx_reduce_kernel<<<(BATCH * EDIM / 4) / 256, 256, 0, stream>>>(ctxp, ctx);

  // combine + GRU
  combine_kernel<<<(BATCH * HDIM) / 256, 256, 0, stream>>>(prev_y, ctx, Wc_w, Wc_b, x);
  linear_kernel<<<(BATCH * 3 * HDIM) / 256, 256, 0, stream>>>(x, W_ih, b_ih, gi, 3 * HDIM, HDIM);
  gate_kernel<<<(BATCH * HDIM) / 256, 256, 0, stream>>>(gi, gh, prev_h, hnew, out + OUT_H);

  // output projection (WMMA, Wout streamed once, NT) + log-softmax
  logits_wmma_kernel<<<VOCAB / 16, 32, 0, stream>>>(hnew, Wout_w, Wout_b, logits);
  logsm_kernel<<<BATCH, 256, 0, stream>>>(logits, out);
}